// LongTermMemory_16801912062742
// MI455X (gfx1250) — compile-verified
//
#include <hip/hip_runtime.h>
#include <hip/hip_bf16.h>
#include <math.h>

// ---------------------------------------------------------------------------
// CDNA5 (gfx1250) fused LongTermMemory pipeline, fp32 WMMA throughout.
// D = A(16x4 f32) x B(4x16 f32) + C(16x16 f32), wave32.
// A frag: lanes 0-15 hold M=lane, v0=K0,v1=K1; lanes 16-31 hold M=lane-16, K2/K3.
// B frag (mirror): lanes 0-15 hold N=lane, v0=K0,v1=K1; lanes 16-31 K2/K3.
// C/D: vgpr i, lanes 0-15 -> (M=i, N=lane); lanes 16-31 -> (M=i+8, N=lane-16).
// ---------------------------------------------------------------------------

typedef __attribute__((ext_vector_type(2))) float v2f;
typedef __attribute__((ext_vector_type(8))) float v8f;

#define INF_F 3.0e38f
#define EPS_F 1e-6f

#define NKEYS  100000
#define DKEY   128
#define NSPLIT 16
#define CHUNK  (NKEYS / NSPLIT)   // 6250

__device__ __forceinline__ void topk5_insert(float d, int id, float td[5], int ti[5]) {
  if (d >= td[4]) return;
  td[4] = d; ti[4] = id;
#pragma unroll
  for (int j = 4; j > 0; --j) {
    if (td[j] < td[j - 1]) {
      float t = td[j]; td[j] = td[j - 1]; td[j - 1] = t;
      int   u = ti[j]; ti[j] = ti[j - 1]; ti[j - 1] = u;
    }
  }
}

// ---------------------------------------------------------------------------
// Key squared-norms, computed once (keys are constant; the old code recomputed
// them 64x inside the hot loop).  One wave per key row, coalesced float4 loads.
// ---------------------------------------------------------------------------
__global__ __launch_bounds__(256) void knorm_kernel(const float* __restrict__ keys,
                                                    float* __restrict__ kn)
{
  const int lane = threadIdx.x & 31;
  const int row  = blockIdx.x * 8 + (threadIdx.x >> 5);
  if (row >= NKEYS) return;
  const float4 v = *(const float4*)(keys + (size_t)row * DKEY + lane * 4);
  float s = v.x * v.x + v.y * v.y + v.z * v.z + v.w * v.w;
#pragma unroll
  for (int m = 16; m >= 1; m >>= 1) s += __shfl_xor(s, m, 32);
  if (lane == 0) kn[row] = s;
}

// ---------------------------------------------------------------------------
// Y[M,N] = act(X[M,K] @ W[K,N] + bias).  One wave per 16x16 tile, 4 waves/block.
// grid = (N/64, M/16), block = 128.  K % 16 == 0.  Loads batched 4 k-steps at
// a time so multiple loads stay in flight per WMMA group.
// ---------------------------------------------------------------------------
template <bool RELU>
__global__ __launch_bounds__(128) void gemm_bias_kernel(
    const float* __restrict__ X, const float* __restrict__ W,
    const float* __restrict__ B, float* __restrict__ Y,
    int M, int N, int K)
{
  const int l      = threadIdx.x & 31;
  const int wave   = threadIdx.x >> 5;
  const int half   = l >> 4;
  const int lane16 = l & 15;
  const int mt = blockIdx.y;
  const int nt = blockIdx.x * 4 + wave;

  const float* xp = X + (size_t)(mt * 16 + lane16) * K + 2 * half;
  const float* wp = W + (size_t)(nt * 16 + lane16) + (size_t)(2 * half) * N;

  v8f acc = {};
  for (int k0 = 0; k0 < K; k0 += 16) {
    v2f a[4], b[4];
#pragma unroll
    for (int u = 0; u < 4; ++u) {
      a[u]   = *(const v2f*)(xp + k0 + 4 * u);
      b[u].x = wp[(size_t)(k0 + 4 * u)     * N];
      b[u].y = wp[(size_t)(k0 + 4 * u + 1) * N];
    }
#pragma unroll
    for (int u = 0; u < 4; ++u)
      acc = __builtin_amdgcn_wmma_f32_16x16x4_f32(false, a[u], false, b[u],
                                                  (short)0, acc, false, false);
  }

  const int   col = nt * 16 + lane16;
  const float bv  = B[col];
#pragma unroll
  for (int i = 0; i < 8; ++i) {
    const int r = mt * 16 + i + 8 * half;
    float y = acc[i] + bv;
    if (RELU) y = fmaxf(y, 0.0f);
    Y[(size_t)r * N + col] = y;
  }
}

// ---------------------------------------------------------------------------
// Fused squared-L2 distance + per-row top-5 (partial).
// grid = (NSPLIT, B/16), block = 256 (8 waves).  Block owns 16 query rows and
// a CHUNK-key split.  The whole 16x128 A operand (cq tile) lives in registers
// (v2f a[32]) -- it is invariant across all key tiles, so the inner loop is
// pure global-load + WMMA.  Key norms come precomputed from knorm[].
// ---------------------------------------------------------------------------
__global__ __launch_bounds__(256) void dist_topk_kernel(
    const float* __restrict__ cq, const float* __restrict__ keys,
    const float* __restrict__ knorm,
    float* __restrict__ pd, int* __restrict__ pi)
{
  __shared__ float tile_s[8][16][17];

  const int tid   = threadIdx.x;
  const int rb    = blockIdx.y;
  const int split = blockIdx.x;
  const int l      = tid & 31;
  const int wave   = tid >> 5;
  const int half   = l >> 4;
  const int lane16 = l & 15;

  // A fragments for the whole K=128 held in registers; row m = lane16.
  const float* qp = cq + (size_t)(rb * 16 + lane16) * DKEY + 2 * half;
  v2f a[32];
  float qnp = 0.f;
#pragma unroll
  for (int t = 0; t < 32; ++t) {
    a[t] = *(const v2f*)(qp + 4 * t);
    qnp += a[t].x * a[t].x + a[t].y * a[t].y;
  }
  const float qnr = qnp + __shfl_xor(qnp, 16, 32);  // ||q_row||^2, row = lane16

  float td[5]; int ti[5];
#pragma unroll
  for (int j = 0; j < 5; ++j) { td[j] = INF_F; ti[j] = -1; }

  const int nIter = (CHUNK + 127) / 128;   // 49
  for (int it = 0; it < nIter; ++it) {
    const int lkBase = it * 128 + wave * 16;      // local key index of col 0
    const int gkBase = split * CHUNK + lkBase;    // global key index of col 0
    int krow = gkBase + lane16;
    if (krow > NKEYS - 1) krow = NKEYS - 1;       // clamp loads; masked later
    const float* kp = keys + (size_t)krow * DKEY + 2 * half;

    // prefetch next tile's key row (L2-resident after first sweep)
    if (it + 1 < nIter) {
      int nrow = krow + 128; if (nrow > NKEYS - 1) nrow = NKEYS - 1;
      __builtin_prefetch(keys + (size_t)nrow * DKEY + 2 * half, 0, 1);
    }

    v8f acc = {};
#pragma unroll
    for (int t8 = 0; t8 < 4; ++t8) {
      v2f b[8];
#pragma unroll
      for (int u = 0; u < 8; ++u)               // 8 loads issued as a clause
        b[u] = *(const v2f*)(kp + (t8 * 8 + u) * 4);
#pragma unroll
      for (int u = 0; u < 8; ++u)               // 8 WMMAs back-to-back
        acc = __builtin_amdgcn_wmma_f32_16x16x4_f32(false, a[t8 * 8 + u], false,
                                                    b[u], (short)0, acc,
                                                    false, false);
    }

#pragma unroll
    for (int i = 0; i < 8; ++i) tile_s[wave][i + 8 * half][lane16] = acc[i];
    __syncthreads();

    // scan: lane owns row r = lane16, columns [half*8, half*8+8)
    const int cbase = half * 8;
#pragma unroll
    for (int j = 0; j < 8; ++j) {
      const int col  = cbase + j;
      const int lk   = lkBase + col;
      const int gidx = gkBase + col;
      const int cidx = gidx > NKEYS - 1 ? NKEYS - 1 : gidx;
      float d = qnr + knorm[cidx] - 2.0f * tile_s[wave][lane16][col];
      if (lk >= CHUNK) d = INF_F;               // mask tail / cross-split keys
      topk5_insert(d, gidx, td, ti);
    }
    __syncthreads();
  }

  // merge with partner lane (same row, other column half)
  float od[5]; int oi[5];
#pragma unroll
  for (int j = 0; j < 5; ++j) {
    od[j] = __shfl_xor(td[j], 16, 32);
    oi[j] = __shfl_xor(ti[j], 16, 32);
  }
#pragma unroll
  for (int j = 0; j < 5; ++j) topk5_insert(od[j], oi[j], td, ti);

  if (l < 16) {
    const size_t base = ((((size_t)rb * NSPLIT + split) * 8 + wave) * 16 + l) * 5;
#pragma unroll
    for (int j = 0; j < 5; ++j) { pd[base + j] = td[j]; pi[base + j] = ti[j]; }
  }
}

// ---------------------------------------------------------------------------
// Merge 640 candidates/row -> top-5, compute weights, gather values,
// write weighted [B,128] and confidence [B].  One wave per row.
// grid = B/8 = 128, block = 256.
// ---------------------------------------------------------------------------
__global__ __launch_bounds__(256) void topk_reduce_kernel(
    const float* __restrict__ pd, const int* __restrict__ pi,
    const float* __restrict__ values, float* __restrict__ wtd,
    float* __restrict__ conf)
{
  const int lane = threadIdx.x & 31;
  const int row  = blockIdx.x * 8 + (threadIdx.x >> 5);
  const int rb = row >> 4, rr = row & 15;

  float td[5]; int ti[5];
#pragma unroll
  for (int j = 0; j < 5; ++j) { td[j] = INF_F; ti[j] = 0; }

  const int NC = NSPLIT * 8 * 5;    // 640 candidates per row
  for (int cc = lane; cc < NC; cc += 32) {
    const int s = cc / 40;
    const int rem = cc - s * 40;
    const int w = rem / 5;
    const int j = rem - w * 5;
    const size_t off = ((((size_t)rb * NSPLIT + s) * 8 + w) * 16 + rr) * 5 + j;
    topk5_insert(pd[off], pi[off], td, ti);
  }
  // butterfly merge across the wave (snapshot partner before inserting)
#pragma unroll
  for (int mask = 16; mask >= 1; mask >>= 1) {
    float od[5]; int oi[5];
#pragma unroll
    for (int j = 0; j < 5; ++j) {
      od[j] = __shfl_xor(td[j], mask, 32);
      oi[j] = __shfl_xor(ti[j], mask, 32);
    }
#pragma unroll
    for (int j = 0; j < 5; ++j) topk5_insert(od[j], oi[j], td, ti);
  }

  float w[5], s = 0.f;
#pragma unroll
  for (int j = 0; j < 5; ++j) { w[j] = 1.0f / (td[j] + EPS_F); s += w[j]; }
  const float inv = 1.0f / s;

  for (int c = lane; c < DKEY; c += 32) {
    float acc = 0.f;
#pragma unroll
    for (int j = 0; j < 5; ++j)
      acc += (w[j] * inv) * values[(size_t)ti[j] * DKEY + c];
    wtd[(size_t)row * DKEY + c] = acc;
  }
  if (lane == 0) conf[row] = 1.0f / (td[0] + EPS_F);
}

// ---------------------------------------------------------------------------
extern "C" void kernel_launch(void* const* d_in, const int* in_sizes, int n_in,
                              void* d_out, int out_size, void* d_ws, size_t ws_size,
                              hipStream_t stream)
{
  const float* query = (const float*)d_in[0];
  const float* W1    = (const float*)d_in[1];
  const float* b1    = (const float*)d_in[2];
  const float* W2    = (const float*)d_in[3];
  const float* b2    = (const float*)d_in[4];
  const float* W3    = (const float*)d_in[5];
  const float* b3    = (const float*)d_in[6];
  const float* keys  = (const float*)d_in[7];
  const float* values= (const float*)d_in[8];
  const float* D1w   = (const float*)d_in[9];
  const float* D1b   = (const float*)d_in[10];
  const float* D2w   = (const float*)d_in[11];
  const float* D2b   = (const float*)d_in[12];
  const float* D3w   = (const float*)d_in[13];
  const float* D3b   = (const float*)d_in[14];
  // d_in[15] = k (always 5 in this problem)

  float* ws  = (float*)d_ws;
  float* h1  = ws;                    // [1024, 512]
  float* h2  = h1 + 1024 * 512;       // [1024, 256]
  float* cq  = h2 + 1024 * 256;       // [1024, 128]
  float* wtd = cq + 1024 * 128;       // [1024, 128]
  float* pd  = wtd + 1024 * 128;      // [64][16][8][16][5] dist
  int*   pi  = (int*)(pd + 64 * NSPLIT * 8 * 16 * 5);
  float* kn  = (float*)(pi + 64 * NSPLIT * 8 * 16 * 5);  // [100000]

  float* out  = (float*)d_out;        // retrieved [1024,1024]
  float* conf = out + 1024 * 1024;    // confidence [1024]

  // Key norms once (independent of query path; overlaps encoder MLP in graph)
  knorm_kernel<<<dim3((NKEYS + 7) / 8), 256, 0, stream>>>(keys, kn);

  // Encoder MLP: query -> cq
  gemm_bias_kernel<true ><<<dim3( 8, 64), 128, 0, stream>>>(query, W1, b1, h1, 1024,  512, 1024);
  gemm_bias_kernel<true ><<<dim3( 4, 64), 128, 0, stream>>>(h1,    W2, b2, h2, 1024,  256,  512);
  gemm_bias_kernel<false><<<dim3( 2, 64), 128, 0, stream>>>(h2,    W3, b3, cq, 1024,  128,  256);

  // Fused distance + partial top-5, then reduce + weighted gather
  dist_topk_kernel<<<dim3(NSPLIT, 64), 256, 0, stream>>>(cq, keys, kn, pd, pi);
  topk_reduce_kernel<<<dim3(128), 256, 0, stream>>>(pd, pi, values, wtd, conf);

  // Decoder MLP: weighted -> retrieved (final layer straight into d_out)
  gemm_bias_kernel<true ><<<dim3( 4, 64), 128, 0, stream>>>(wtd, D1w, D1b, h2, 1024,  256,  128);
  gemm_bias_kernel<true ><<<dim3( 8, 64), 128, 0, stream>>>(h2,  D2w, D2b, h1, 1024,  512,  256);
  gemm_bias_kernel<false><<<dim3(16, 64), 128, 0, stream>>>(h1,  D3w, D3b, out, 1024, 1024,  512);
}